// RecLoss_49804440764587
// MI455X (gfx1250) — compile-verified
//
#include <hip/hip_runtime.h>

#define N_USERS 200000
#define N_ITEMS 100000
#define NREL 3
#define DIM 64
#define BATCH 8192
#define LLEN 100
#define ROWF 192          // floats per (user|item) row: NREL*DIM
#define CHUNK 128         // rows staged in LDS per chunk
#define LDS_STRIDE 72     // padded row stride (floats): 288B, 16B-aligned, banks spread

// workspace float offsets
#define GII 0             // 3 * 4096 item grams
#define GUU (3*4096)      // 3 * 4096 user grams
#define POS (6*4096)      // pos-term accumulator (coef folded in)
#define WS_FLOATS (POS + 1)

typedef __attribute__((ext_vector_type(2))) float v2f;
typedef __attribute__((ext_vector_type(8))) float v8f;

__global__ __launch_bounds__(256) void zero_ws_kernel(float* ws, int n) {
    int i = blockIdx.x * blockDim.x + threadIdx.x;
    if (i < n) ws[i] = 0.0f;
}

// Issue 8 async B128 copies per lane: CHUNK x 64 floats -> LDS buffer.
// Rows clamped to [0, nrows-1] so every global address is valid; tail rows
// are zero-fixed after the wait. Exactly 8 ASYNCcnt increments per wave.
__device__ __forceinline__ void async_stage(const float* __restrict__ src,
                                            const int* __restrict__ gather,
                                            int nrows, int rel, int base,
                                            float* ldsBuf, int tid) {
    #pragma unroll
    for (int it = 0; it < 8; ++it) {
        const int j  = tid + it * 256;       // float4 index within chunk (0..2047)
        const int r  = j >> 4;
        const int c4 = (j & 15) << 2;
        int row = base + r;
        row = (row < nrows) ? row : (nrows - 1);
        const int srow = gather ? gather[row] : row;
        const float* g = src + (size_t)srow * ROWF + rel * DIM + c4;
        const unsigned lds =
            (unsigned)(unsigned long long)(ldsBuf + r * LDS_STRIDE + c4);
        asm volatile("global_load_async_to_lds_b128 %0, %1, off"
                     :: "v"(lds), "v"(g) : "memory");
    }
}

// G[rel] += X_cols(rel)^T * X_cols(rel) over nrows rows of a 192-float-stride
// matrix (optionally gathered). Double-buffered async LDS staging + fp32 WMMA.
__global__ __launch_bounds__(256)
void gram_kernel(const float* __restrict__ src, const int* __restrict__ gather,
                 int nrows, float* __restrict__ gramBase) {
    __shared__ __align__(16) float Xs[2][CHUNK * LDS_STRIDE];

    const int tid  = threadIdx.x;
    const int rel  = blockIdx.y;
    const int lane = tid & 31;
    const int wave = tid >> 5;
    const int m    = lane & 15;
    const int h    = lane >> 4;

    // two 16x16 tiles per wave: t and t+8 of the 4x4 tile grid
    const int t0 = wave, t1 = wave + 8;
    const int bi0 = t0 >> 2, bj0 = t0 & 3;
    const int bi1 = t1 >> 2, bj1 = t1 & 3;

    v8f c0 = {}; v8f c1 = {};

    const int nChunks = (nrows + CHUNK - 1) / CHUNK;
    int ch  = blockIdx.x;
    int buf = 0;
    if (ch < nChunks) async_stage(src, gather, nrows, rel, ch * CHUNK, Xs[0], tid);

    for (; ch < nChunks; ch += gridDim.x) {
        const int next = ch + gridDim.x;
        const bool haveNext = next < nChunks;          // block-uniform
        if (haveNext)
            async_stage(src, gather, nrows, rel, next * CHUNK, Xs[buf ^ 1], tid);
        // async loads complete in order: <=8 outstanding => current buffer done
        if (haveNext) asm volatile("s_wait_asynccnt 0x8" ::: "memory");
        else          asm volatile("s_wait_asynccnt 0x0" ::: "memory");
        __syncthreads();

        const int base = ch * CHUNK;
        float* Xb = &Xs[buf][0];
        if (base + CHUNK > nrows) {                    // block-uniform tail fix
            for (int i = tid; i < CHUNK * DIM; i += 256) {
                const int r = i >> 6;
                if (base + r >= nrows) Xb[r * LDS_STRIDE + (i & 63)] = 0.0f;
            }
            __syncthreads();
        }

        // A[m,k] = X[k][bi*16+m], B[k,n] = X[k][bj*16+n]; fp32 16x4 layout:
        // lanes 0-15 hold K={k0,k0+1}, lanes 16-31 hold K={k0+2,k0+3}
        for (int k0 = 0; k0 < CHUNK; k0 += 4) {
            const int kA = k0 + 2 * h;
            v2f a0, b0, a1, b1;
            a0.x = Xb[kA * LDS_STRIDE + bi0 * 16 + m];
            a0.y = Xb[(kA + 1) * LDS_STRIDE + bi0 * 16 + m];
            b0.x = Xb[kA * LDS_STRIDE + bj0 * 16 + m];
            b0.y = Xb[(kA + 1) * LDS_STRIDE + bj0 * 16 + m];
            c0 = __builtin_amdgcn_wmma_f32_16x16x4_f32(false, a0, false, b0,
                                                       (short)0, c0, false, false);
            a1.x = Xb[kA * LDS_STRIDE + bi1 * 16 + m];
            a1.y = Xb[(kA + 1) * LDS_STRIDE + bi1 * 16 + m];
            b1.x = Xb[kA * LDS_STRIDE + bj1 * 16 + m];
            b1.y = Xb[(kA + 1) * LDS_STRIDE + bj1 * 16 + m];
            c1 = __builtin_amdgcn_wmma_f32_16x16x4_f32(false, a1, false, b1,
                                                       (short)0, c1, false, false);
        }
        buf ^= 1;
        __syncthreads();   // all readers done before next stage overwrites
    }

    // D layout: VGPR v -> (M = v + 8*h, N = m)
    float* g = gramBase + rel * (DIM * DIM);
    #pragma unroll
    for (int v = 0; v < 8; ++v) {
        const int rrow = v + 8 * h;
        atomicAdd(&g[(bi0 * 16 + rrow) * DIM + bj0 * 16 + m], c0[v]);
        atomicAdd(&g[(bi1 * 16 + rrow) * DIM + bj1 * 16 + m], c1[v]);
    }
}

// Per batch-row gathered dot products: sum_rel coef[rel]*((1-W)*p^2 - 2p)
__global__ __launch_bounds__(128)
void pos_kernel(const int* __restrict__ input_u,
                const int* __restrict__ lab0, const int* __restrict__ lab1,
                const int* __restrict__ lab2,
                const float* __restrict__ ua, const float* __restrict__ ia,
                const float* __restrict__ r0, const float* __restrict__ r1,
                const float* __restrict__ r2, float* __restrict__ posAcc) {
    __shared__ float uiL[ROWF];
    __shared__ float rL[ROWF];
    __shared__ float red[128];

    const int a   = blockIdx.x;
    const int tid = threadIdx.x;
    const int user = input_u[a];
    for (int j = tid; j < ROWF; j += 128) {
        uiL[j] = ua[(size_t)user * ROWF + j];
        rL[j]  = (j < 64) ? r0[j] : (j < 128 ? r1[j - 64] : r2[j - 128]);
    }
    __syncthreads();

    const int* labs[NREL] = {lab0, lab1, lab2};
    const float coef[NREL] = {0.16666f, 0.66666f, 0.16666f};
    const float omw = 1.0f - 0.1f;   // 1 - WID

    float acc = 0.0f;
    for (int p = tid; p < NREL * LLEN; p += 128) {
        const int rel = p / LLEN;
        const int l   = p - rel * LLEN;
        const int lab = labs[rel][(size_t)a * LLEN + l];
        if (lab != N_ITEMS) {
            const float* ii = ia + ((size_t)lab * NREL + rel) * DIM;
            const float* u  = uiL + rel * DIM;
            const float* rv = rL + rel * DIM;
            float s = 0.0f;
            #pragma unroll
            for (int k = 0; k < DIM; k += 4) {
                const float4 iv = *(const float4*)(ii + k);
                s += iv.x * u[k] * rv[k] + iv.y * u[k + 1] * rv[k + 1] +
                     iv.z * u[k + 2] * rv[k + 2] + iv.w * u[k + 3] * rv[k + 3];
            }
            acc += coef[rel] * (omw * s * s - 2.0f * s);
        }
    }

    red[tid] = acc;
    __syncthreads();
    for (int s = 64; s > 0; s >>= 1) {
        if (tid < s) red[tid] += red[tid + s];
        __syncthreads();
    }
    if (tid == 0) atomicAdd(posAcc, red[0]);
}

__device__ __forceinline__ float block_reduce256(float v, float* red, int tid) {
    red[tid] = v;
    __syncthreads();
    for (int s = 128; s > 0; s >>= 1) {
        if (tid < s) red[tid] += red[tid + s];
        __syncthreads();
    }
    const float r = red[0];
    __syncthreads();
    return r;
}

__global__ __launch_bounds__(256)
void finalize_kernel(const float* __restrict__ ws, const float* __restrict__ r0,
                     const float* __restrict__ r1, const float* __restrict__ r2,
                     float* __restrict__ out) {
    __shared__ float red[256];
    const int tid = threadIdx.x;
    const float* rs[NREL] = {r0, r1, r2};
    const float coef[NREL] = {0.16666f, 0.66666f, 0.16666f};
    const float wid = 0.1f;

    float accLoss = 0.0f, trI = 0.0f, trU = 0.0f;
    for (int rel = 0; rel < NREL; ++rel) {
        const float* gii = ws + GII + rel * (DIM * DIM);
        const float* guu = ws + GUU + rel * (DIM * DIM);
        const float* rv  = rs[rel];
        for (int idx = tid; idx < DIM * DIM; idx += 256) {
            const int b = idx >> 6, c = idx & 63;
            const float gi = gii[idx], gu = guu[idx];
            accLoss += coef[rel] * wid * gi * gu * rv[b] * rv[c];
            if (b == c) { trI += gi; trU += gu; }   // traces = sums of squares
        }
    }
    const float lossGram = block_reduce256(accLoss, red, tid);
    const float trIs     = block_reduce256(trI, red, tid);
    const float trUs     = block_reduce256(trU, red, tid);
    if (tid == 0) {
        out[0] = lossGram + ws[POS];               // loss
        out[1] = 0.01f * 0.5f * (trIs + trUs);     // emb_loss = DECAY * regularizer
    }
}

extern "C" void kernel_launch(void* const* d_in, const int* in_sizes, int n_in,
                              void* d_out, int out_size, void* d_ws, size_t ws_size,
                              hipStream_t stream) {
    const int*   input_u = (const int*)d_in[0];
    const int*   lab0    = (const int*)d_in[1];
    const int*   lab1    = (const int*)d_in[2];
    const int*   lab2    = (const int*)d_in[3];
    const float* ua      = (const float*)d_in[4];
    const float* ia      = (const float*)d_in[5];
    const float* r0      = (const float*)d_in[6];
    const float* r1      = (const float*)d_in[7];
    const float* r2      = (const float*)d_in[8];
    float* ws  = (float*)d_ws;
    float* out = (float*)d_out;

    zero_ws_kernel<<<(WS_FLOATS + 255) / 256, 256, 0, stream>>>(ws, WS_FLOATS);

    dim3 gItems(96, NREL);
    gram_kernel<<<gItems, 256, 0, stream>>>(ia, nullptr, N_ITEMS + 1, ws + GII);
    dim3 gUsers(16, NREL);
    gram_kernel<<<gUsers, 256, 0, stream>>>(ua, input_u, BATCH, ws + GUU);

    pos_kernel<<<BATCH, 128, 0, stream>>>(input_u, lab0, lab1, lab2, ua, ia,
                                          r0, r1, r2, &ws[POS]);

    finalize_kernel<<<1, 256, 0, stream>>>(ws, r0, r1, r2, out);
}